// P2SP_50508815401493
// MI455X (gfx1250) — compile-verified
//
#include <hip/hip_runtime.h>
#include <hip/hip_bf16.h>

typedef __attribute__((ext_vector_type(2))) float v2f;
typedef __attribute__((ext_vector_type(8))) float v8f;

namespace {
constexpr int Bsz = 16, Csz = 16, Hsz = 512, Wsz = 512;
constexpr int HC = 16, WC = 16;          // cell (stoken) size
constexpr int HH = Hsz / HC;             // 32 cells in y
constexpr int WW = Wsz / WC;             // 32 cells in x
constexpr int HW = Hsz * Wsz;            // channel stride in x / plane stride in out
}

// One workgroup (256 threads = 8 waves) per cell (b, cy, cx).
// Phase 1: stage sf[16 ch][16 cols] in LDS (cols 9..15 zero -> clean B pad).
// Phase 2: each wave computes two 16x16 M-tiles of aff = pf(256x16) @ sf(16x9)
//          via 4 chained V_WMMA_F32_16X16X4_F32 each, spills D to LDS.
// Phase 3: thread p does scale*para + softmax over k=0..8, coalesced store.
__global__ __launch_bounds__(256) void p2sp_wmma_kernel(
    const float* __restrict__ x,
    const float* __restrict__ st,
    const float* __restrict__ para,
    float* __restrict__ out)
{
  __shared__ float sf[Csz][16];        // [channel][neighbor k], k>=9 zeroed
  __shared__ float dmat[16][16][17];   // [M-tile][row m][col k], padded stride

  const int tid  = threadIdx.x;        // 0..255
  const int lane = tid & 31;
  const int wave = tid >> 5;           // 0..7
  const int hl   = lane & 15;          // lane within half-wave
  const int half = lane >> 4;          // 0/1: which K-pair (A) / K-row pair (B)

  const int cell = blockIdx.x;
  const int cx = cell % WW;
  const int cy = (cell / WW) % HH;
  const int b  = cell / (WW * HH);

  // ---- Phase 1: stage B-matrix sf[c][k] = stoken(b, c, cy+dy-1, cx+dx-1), 0-pad
  {
    const int c = tid >> 4;            // 0..15
    const int k = tid & 15;            // 0..15 (9..15 -> zero pad columns)
    float v = 0.0f;
    if (k < 9) {
      const int iy = cy + (k / 3) - 1;
      const int ix = cx + (k % 3) - 1;
      if (iy >= 0 && iy < HH && ix >= 0 && ix < WW)
        v = st[((size_t)(b * Csz + c) * HH + iy) * WW + ix];
    }
    sf[c][k] = v;
  }
  __syncthreads();

  // Base of this cell's pixel block in x: pf[p, c] = xb[c*HW + (p/16)*W + (p%16)]
  const float* xb = x + (size_t)b * Csz * HW + (size_t)(cy * HC) * Wsz + cx * WC;

  // ---- Phase 2: WMMA GEMM, each wave handles M-tiles (wave*2) and (wave*2+1)
  #pragma unroll
  for (int ti = 0; ti < 2; ++ti) {
    const int tile = wave * 2 + ti;    // pixel rows p = tile*16 + m
    v8f acc = {};                      // C/D accumulator, starts at 0
    #pragma unroll
    for (int k0 = 0; k0 < Csz; k0 += 4) {
      // A 16x4 f32 layout: lanes 0-15 hold K=k0,k0+1; lanes 16-31 K=k0+2,k0+3
      const int cA = k0 + half * 2;
      const float* ap = xb + (size_t)cA * HW + tile * Wsz + hl; // 64B/half-wave
      v2f a, bf;
      a.x = ap[0];                     // pf[p, cA]
      a.y = ap[HW];                    // pf[p, cA+1]
      // B 4x16 f32 layout: VGPR0/1 rows K=cA,cA+1; N = hl (cols 9..15 are zero)
      bf.x = sf[cA][hl];
      bf.y = sf[cA + 1][hl];
      acc = __builtin_amdgcn_wmma_f32_16x16x4_f32(
          /*neg_a=*/false, a, /*neg_b=*/false, bf,
          /*c_mod=*/(short)0, acc, /*reuse_a=*/false, /*reuse_b=*/false);
    }
    // D layout: VGPR r = row M=r (lanes 0-15) / M=r+8 (lanes 16-31), N = hl
    #pragma unroll
    for (int r = 0; r < 8; ++r)
      dmat[tile][r + half * 8][hl] = acc[r];
  }
  __syncthreads();

  // ---- Phase 3: scale * para, softmax over 9 neighbors, scatter store
  const int p  = tid;                  // pixel index within cell, 0..255
  const int mt = p >> 4;               // M-tile
  const int m  = p & 15;               // row within tile
  const float scale = 0.25f;           // C^-0.5, C=16

  float v[9];
  float mx = -3.402823466e38f;
  #pragma unroll
  for (int k = 0; k < 9; ++k) {
    const float t = dmat[mt][m][k] * scale * para[p * 9 + k];
    v[k] = t;
    mx = fmaxf(mx, t);
  }
  float s = 0.0f;
  #pragma unroll
  for (int k = 0; k < 9; ++k) {
    const float e = __expf(v[k] - mx);
    v[k] = e;
    s += e;
  }
  const float inv = 1.0f / s;

  // out[b, k, cy*16 + p/16, cx*16 + p%16]; consecutive threads -> consecutive px
  const int py = p >> 4, px = p & 15;
  float* ob = out + (size_t)b * 9 * HW + (size_t)(cy * HC + py) * Wsz
            + (cx * WC + px);
  #pragma unroll
  for (int k = 0; k < 9; ++k)
    ob[(size_t)k * HW] = v[k] * inv;
}

extern "C" void kernel_launch(void* const* d_in, const int* in_sizes, int n_in,
                              void* d_out, int out_size, void* d_ws, size_t ws_size,
                              hipStream_t stream) {
  const float* x    = (const float*)d_in[0];  // [16,16,512,512] f32
  const float* st   = (const float*)d_in[1];  // [16,16,32,32]   f32
  const float* para = (const float*)d_in[2];  // [1,1,256,9]     f32
  float* out = (float*)d_out;                 // [16,9,512,512]  f32

  dim3 grid(Bsz * HH * WW);   // 16384 cells
  dim3 block(256);            // 8 waves of 32
  p2sp_wmma_kernel<<<grid, block, 0, stream>>>(x, st, para, out);
}